// TwoLayerNet_12945031430555
// MI455X (gfx1250) — compile-verified
//
#include <hip/hip_runtime.h>

typedef float v2f __attribute__((ext_vector_type(2)));
typedef float v8f __attribute__((ext_vector_type(8)));
typedef unsigned int u32x4 __attribute__((ext_vector_type(4)));
typedef int i32x4 __attribute__((ext_vector_type(4)));
typedef int i32x8 __attribute__((ext_vector_type(8)));

#define KM_PITCH 68   // 64 + 4 pad: row and column A-frag reads both conflict-free
#define ST_PITCH 34   // 32 + 2 pad: C-tile scatter conflict-free
#define U_PITCH 136   // 128 + 8 pad: 2-row B-frag lane split maps to disjoint banks

__device__ __forceinline__ v8f wmma16x16x4(v2f a, v2f b, v8f c) {
  // D = A(16x4,f32) x B(4x16,f32) + C(16x16,f32) -- native fp32 WMMA
  return __builtin_amdgcn_wmma_f32_16x16x4_f32(false, a, false, b, (short)0, c,
                                               false, false);
}

// ---- Tensor Data Mover: DMA a 64-row x 128-col fp32 block (row-contiguous in
// global) into LDS at lds_off, inserting 8 DWORDs of padding after every 128
// DWORDs so the LDS image has pitch U_PITCH (=136 floats). D# bitfields per
// CDNA5 ISA 8.3/8.4 (group0: count/lds/global/type, group1: dims/strides/pad).
__device__ __forceinline__ void tdm_load_block(const void* gptr, unsigned lds_off) {
  unsigned long long ga = (unsigned long long)(uintptr_t)gptr;
  u32x4 g0;
  g0.x = 1u;                                   // count=1 (valid user descriptor)
  g0.y = lds_off;                              // lds_addr (bytes)
  g0.z = (unsigned)(ga & 0xffffffffu);         // global_addr[31:0]
  g0.w = (unsigned)((ga >> 32) & 0x01ffffffu)  // global_addr[56:32]
         | (2u << 30);                         // type=2 ("image")
  i32x8 g1;
  g1[0] = (2 << 16)      // data_size = 4 bytes
        | (1 << 20)      // pad_enable
        | (6 << 22)      // pad_interval: 128 DWORDs
        | (7 << 25);     // pad_amount: 8 DWORDs
  g1[1] = (128 << 16);   // tensor_dim0 = 128 (low 16 bits at [63:48])
  g1[2] = (64 << 16);    // tensor_dim0 hi16=0 | tensor_dim1 = 64 (low16 at [95:80])
  g1[3] = (128 << 16);   // tensor_dim1 hi16=0 | tile_dim0 = 128
  g1[4] = 64;            // tile_dim1 = 64 | tile_dim2 = 0
  g1[5] = 128;           // tensor_dim0_stride = 128 (low 32)
  g1[6] = 0;             // stride hi | tensor_dim1_stride lo
  g1[7] = 0;
  i32x4 z4 = {0, 0, 0, 0};
#if __clang_major__ >= 23
  i32x8 z8 = {0, 0, 0, 0, 0, 0, 0, 0};
  __builtin_amdgcn_tensor_load_to_lds(g0, g1, z4, z4, z8, 0);
#else
  __builtin_amdgcn_tensor_load_to_lds(g0, g1, z4, z4, 0);
#endif
}

// ---------------- Kernel 1: batched fixed-point solver ----------------
// 512 threads = 16 wave32, 32 batch columns per block.
// Chain A (waves 0-7):  AF -> Ldi -> AF'   Chain B (waves 8-15): LF -> Adi -> LF'
// Km A-fragments are hoisted out of the iteration loop (Km is constant), so the
// steady-state step is: 16 x { ds B-frag pair ; v_wmma_f32_16x16x4_f32 }.
__global__ __launch_bounds__(512)
void fp_solver_kernel(const float* __restrict__ LT, const float* __restrict__ Kmat,
                      const float* __restrict__ AT, const int* __restrict__ max_step,
                      float* __restrict__ wsAF, float* __restrict__ wsLF) {
  __shared__ float sKm[64][KM_PITCH];   // relu(K), row-major [l][a]
  __shared__ float sAT[64];             // relu(AT)
  __shared__ float sLT[64][ST_PITCH];   // LT^T tile: [l][bcol]
  __shared__ float sAF[64][ST_PITCH];
  __shared__ float sLF[64][ST_PITCH];
  __shared__ float sLdi[64][ST_PITCH];
  __shared__ float sAdi[64][ST_PITCH];

  const int tid = threadIdx.x;
  const int bbase = blockIdx.x * 32;

  for (int i = tid; i < 64 * 64; i += 512) {
    float v = Kmat[i];
    sKm[i >> 6][i & 63] = v > 0.f ? v : 0.f;
  }
  if (tid < 64) {
    float v = AT[tid];
    sAT[tid] = v > 0.f ? v : 0.f;
  }
  for (int i = tid; i < 64 * 32; i += 512) {
    const int bcol = i >> 6, l = i & 63;  // consecutive tid -> consecutive l: coalesced
    sLT[l][bcol] = LT[(size_t)(bbase + bcol) * 64 + l];
    sAF[l][bcol] = 0.f;
    sLF[l][bcol] = 0.f;
  }
  __syncthreads();

  const int wave = tid >> 5;               // wave32
  const int lane = tid & 31;
  const int chain = wave >> 3;             // 0: AF chain, 1: LF chain
  const int m0 = ((wave >> 1) & 3) << 4;   // output row tile 0/16/32/48
  const int n0 = (wave & 1) << 4;          // batch-col tile 0/16
  const int lm = lane & 15;
  const int kh = (lane >> 4) << 1;         // lanes 0-15 -> K{k,k+1}, 16-31 -> K{k+2,k+3}
  const int rb = m0 + ((lane >> 4) << 3);  // C/D tile: 8 rows per lane half
  const int c = n0 + lm;

  // Branchless per-wave addressing: A-frag element pair = km[base + kk*mul],
  // km[base + kk*mul + mul]; mul=1 reads Km rows, mul=KM_PITCH reads Km columns
  // (i.e. Km^T rows). Phase1 chain0 / phase2 chain1 use Km; the others Km^T.
  const float* kmf = &sKm[0][0];
  const int a1Mul = (chain == 0) ? 1 : KM_PITCH;
  const int a1Base = (chain == 0) ? ((m0 + lm) * KM_PITCH + kh) : (kh * KM_PITCH + m0 + lm);
  const int a2Mul = (chain == 0) ? KM_PITCH : 1;
  const int a2Base = (chain == 0) ? (kh * KM_PITCH + m0 + lm) : ((m0 + lm) * KM_PITCH + kh);

  v2f aP1[16], aP2[16];
#pragma unroll
  for (int s = 0; s < 16; ++s) {
    const int o1 = a1Base + 4 * s * a1Mul;
    aP1[s].x = kmf[o1];
    aP1[s].y = kmf[o1 + a1Mul];
    const int o2 = a2Base + 4 * s * a2Mul;
    aP2[s].x = kmf[o2];
    aP2[s].y = kmf[o2 + a2Mul];
  }

  // B operands / destinations / numerators per phase (branchless in the loop).
  const float* b1 = (chain == 0) ? &sAF[0][0] : &sLF[0][0];
  const float* b2 = (chain == 0) ? &sLdi[0][0] : &sAdi[0][0];
  float* d1 = ((chain == 0) ? &sLdi[0][0] : &sAdi[0][0]) + c;
  float* d2 = ((chain == 0) ? &sAF[0][0] : &sLF[0][0]) + c;
  const int bBase = kh * ST_PITCH + n0 + lm;

  float num1[8], num2[8];
#pragma unroll
  for (int r = 0; r < 8; ++r) {
    const float lt = sLT[rb + r][c];
    const float at = sAT[rb + r];
    num1[r] = (chain == 0) ? lt : at;
    num2[r] = (chain == 0) ? at : lt;
  }

  const int nIter = max_step[0] + 1;  // loop + post-loop recompute

  for (int it = 0; it < nIter; ++it) {
    // Phase 1: Ldi = LT*rcp(Km@AF + 1) ; Adi = AT*rcp(KmT@LF + 1)
    {
      v8f acc = {0.f, 0.f, 0.f, 0.f, 0.f, 0.f, 0.f, 0.f};
#pragma unroll
      for (int s = 0; s < 16; ++s) {
        const int bo = bBase + 4 * s * ST_PITCH;
        v2f bf;
        bf.x = b1[bo];
        bf.y = b1[bo + ST_PITCH];
        acc = wmma16x16x4(aP1[s], bf, acc);
      }
#pragma unroll
      for (int r = 0; r < 8; ++r)
        d1[(rb + r) * ST_PITCH] = num1[r] * __builtin_amdgcn_rcpf(acc[r] + 1.f);
    }
    __syncthreads();
    // Phase 2: AF = AT*rcp(KmT@Ldi + 1) ; LF = LT*rcp(Km@Adi + 1)
    {
      v8f acc = {0.f, 0.f, 0.f, 0.f, 0.f, 0.f, 0.f, 0.f};
#pragma unroll
      for (int s = 0; s < 16; ++s) {
        const int bo = bBase + 4 * s * ST_PITCH;
        v2f bf;
        bf.x = b2[bo];
        bf.y = b2[bo + ST_PITCH];
        acc = wmma16x16x4(aP2[s], bf, acc);
      }
#pragma unroll
      for (int r = 0; r < 8; ++r)
        d2[(rb + r) * ST_PITCH] = num2[r] * __builtin_amdgcn_rcpf(acc[r] + 1.f);
    }
    __syncthreads();
  }

  // Emit AFf (B,64) / LFf (B,64), coalesced over the fast index.
  for (int i = tid; i < 64 * 32; i += 512) {
    const int bcol = i >> 6, r = i & 63;
    wsAF[(size_t)(bbase + bcol) * 64 + r] = sAF[r][bcol];
    wsLF[(size_t)(bbase + bcol) * 64 + r] = sLF[r][bcol];
  }
}

// ---------------- Kernel 2: y = D@u + AF@e + b ----------------
// D[b, l*64+a] = Km[l,a]*LFf[b,l]*AFf[b,a], built per K-block of 64.
// 256 threads (8 waves) -> 16 batch rows x 128 y-cols; wave w owns y-tile w.
// Double-buffered TDM pipeline: DMA of block l+1 overlaps the 128 WMMAs of
// block l; tensor ops complete in order per wave, so s_wait_tensorcnt 1 after
// issuing the prefetch guarantees the current block has landed.
__global__ __launch_bounds__(256)
void out_gemm_kernel(const float* __restrict__ Kmat, const float* __restrict__ u,
                     const float* __restrict__ e, const float* __restrict__ bias,
                     const float* __restrict__ wsAF, const float* __restrict__ wsLF,
                     float* __restrict__ out) {
  __shared__ float sU[2][64][U_PITCH];  // TDM-staged u/e K-blocks (double buffer)
  __shared__ float sD[16][KM_PITCH];    // A-matrix tile: [m][a]
  __shared__ float sAF[16][64];
  __shared__ float sLF[16][64];

  const int tid = threadIdx.x;
  const int bbase = blockIdx.x * 16;

  for (int i = tid; i < 16 * 64; i += 256) {
    const int m = i >> 6, a = i & 63;
    sAF[m][a] = wsAF[(size_t)(bbase + m) * 64 + a];
    sLF[m][a] = wsLF[(size_t)(bbase + m) * 64 + a];
  }

  const int wave = tid >> 5;
  const int lane = tid & 31;
  const int n0 = wave << 4;
  const int lm = lane & 15;
  const int kh = (lane >> 4) << 1;

  // Generic LDS pointers carry the LDS byte offset in their low 32 bits
  // (ISA 10.2); that is what D#.lds_addr wants.
  const unsigned lds0 = (unsigned)(uintptr_t)&sU[0][0][0];
  const unsigned lds1 = (unsigned)(uintptr_t)&sU[1][0][0];

  if (wave == 0) tdm_load_block(u, lds0);  // pre-issue block 0

  v8f acc = {0.f, 0.f, 0.f, 0.f, 0.f, 0.f, 0.f, 0.f};

  for (int l = 0; l < 65; ++l) {  // 64 u-blocks + 1 e-block, same accumulators
    __syncthreads();  // reads of buffer (l+1)&1 (iter l-1) and of sD are done
    if (wave == 0 && l < 64)  // prefetch block l+1 into the other buffer
      tdm_load_block((l + 1 < 64) ? (const void*)(u + (size_t)(l + 1) * 64 * 128)
                                  : (const void*)e,
                     (l & 1) ? lds0 : lds1);
    // Build the D-tile for block l while the DMA(s) are in flight.
    if (l < 64) {
      for (int i = tid; i < 1024; i += 256) {
        const int m = i >> 6, a = i & 63;
        float kv = Kmat[l * 64 + a];  // relu(K) row, L0/L2 resident
        kv = kv > 0.f ? kv : 0.f;
        sD[m][a] = sLF[m][l] * sAF[m][a] * kv;
      }
    } else {
      for (int i = tid; i < 1024; i += 256) {
        const int m = i >> 6, a = i & 63;
        sD[m][a] = sAF[m][a];
      }
    }
    if (wave == 0) {
      if (l < 64)
        __builtin_amdgcn_s_wait_tensorcnt(1);  // block l done; l+1 may fly
      else
        __builtin_amdgcn_s_wait_tensorcnt(0);  // final block done
    }
    __syncthreads();
    const float(*ub)[U_PITCH] = sU[l & 1];
#pragma unroll
    for (int kk = 0; kk < 64; kk += 4) {
      const int k0 = kk + kh;
      v2f af, bf;
      af.x = sD[lm][k0];
      af.y = sD[lm][k0 + 1];
      bf.x = ub[k0][n0 + lm];
      bf.y = ub[k0 + 1][n0 + lm];
      acc = wmma16x16x4(af, bf, acc);
    }
  }

  const float bv = bias[n0 + lm];
  const int rbo = (lane >> 4) << 3;
#pragma unroll
  for (int r = 0; r < 8; ++r)
    out[(size_t)(bbase + rbo + r) * 128 + n0 + lm] = acc[r] + bv;
}

extern "C" void kernel_launch(void* const* d_in, const int* in_sizes, int n_in,
                              void* d_out, int out_size, void* d_ws, size_t ws_size,
                              hipStream_t stream) {
  const float* LT = (const float*)d_in[0];    // (B, 64)
  const float* K = (const float*)d_in[1];     // (64, 64)
  const float* AT = (const float*)d_in[2];    // (64,)
  const float* u = (const float*)d_in[3];     // (4096, 128)
  const float* e = (const float*)d_in[4];     // (64, 128)
  const float* b = (const float*)d_in[5];     // (128,)
  const int* max_step = (const int*)d_in[6];  // scalar
  float* out = (float*)d_out;                 // (B, 128)

  const int B = in_sizes[0] / 64;             // 4096
  float* wsAF = (float*)d_ws;                 // (B, 64)
  float* wsLF = wsAF + (size_t)B * 64;        // (B, 64)

  fp_solver_kernel<<<B / 32, 512, 0, stream>>>(LT, K, AT, max_step, wsAF, wsLF);
  out_gemm_kernel<<<B / 16, 256, 0, stream>>>(K, u, e, b, wsAF, wsLF, out);
}